// CholeskyMDNhead_62869731279571
// MI455X (gfx1250) — compile-verified
//
#include <hip/hip_runtime.h>

#define BB 4
#define NN 207
#define TT 12
#define KK 4
#define DD 2484   // NN*TT
#define DP 2496   // padded to 156*16
#define NB 156    // DP/16

typedef __attribute__((ext_vector_type(2))) float v2f;
typedef __attribute__((ext_vector_type(8))) float v8f;

__device__ __forceinline__ v8f wmma_f32_k4(v2f a, v2f b, v8f c) {
  // D = A(16x4) * B(4x16) + C(16x16), fp32. F32 WMMA has no A/B NEG modifier
  // (NEG = CNeg only per ISA 7.12) -> negate A at fragment load instead.
  return __builtin_amdgcn_wmma_f32_16x16x4_f32(false, a, false, b, (short)0, c,
                                               false, false);
}

__device__ __forceinline__ float wave_reduce_sum(float v) {
#pragma unroll
  for (int off = 16; off > 0; off >>= 1) v += __shfl_xor(v, off, 32);
  return v;
}

__device__ __forceinline__ float block_reduce_sum(float v, float* red) {
  const int tid = threadIdx.x;
  v = wave_reduce_sum(v);
  __syncthreads();
  if ((tid & 31) == 0) red[tid >> 5] = v;
  __syncthreads();
  float r = 0.0f;
  if (tid < (int)(blockDim.x >> 5)) r = red[tid];
  if (tid < 32) r = wave_reduce_sum(r);
  if (tid == 0) red[0] = r;
  __syncthreads();
  return red[0];
}

// ---------------------------------------------------------------------------
// 0) Pad cov -> Apad (identity in pad block), diff = y - mu, zero reg accum.
// ---------------------------------------------------------------------------
__global__ void __launch_bounds__(256)
prep_kernel(const float* __restrict__ cov, const float* __restrict__ y,
            const float* __restrict__ mu, float* __restrict__ Apad,
            float* __restrict__ diff, float* __restrict__ regsum) {
  const size_t total = (size_t)BB * DP * DP;
  const size_t stride = (size_t)gridDim.x * blockDim.x;
  const size_t g = (size_t)blockIdx.x * blockDim.x + threadIdx.x;
  for (size_t idx = g; idx < total; idx += stride) {
    size_t b = idx / ((size_t)DP * DP);
    size_t rem = idx % ((size_t)DP * DP);
    size_t r = rem / DP, c = rem % DP;
    float v;
    if (r < DD && c < DD)
      v = cov[((size_t)b * DD + r) * DD + c];
    else
      v = (r == c) ? 1.0f : 0.0f;
    Apad[idx] = v;
  }
  for (size_t idx = g; idx < (size_t)BB * DD; idx += stride)
    diff[idx] = y[idx] - mu[idx];
  if (g == 0) regsum[0] = 0.0f;
}

// ---------------------------------------------------------------------------
// 1a) Cholesky step: factor the 16x16 diagonal block of panel jb.
// ---------------------------------------------------------------------------
__global__ void __launch_bounds__(256)
chol_diag_kernel(float* __restrict__ A, int jb) {
  const int b = blockIdx.x;
  float* Ab = A + (size_t)b * DP * DP;
  __shared__ float diag[16 * 17];
  const int tid = threadIdx.x;
  const int j0 = jb * 16;
  {
    int r = tid >> 4, c = tid & 15;
    diag[r * 17 + c] = Ab[(size_t)(j0 + r) * DP + (j0 + c)];
  }
  __syncthreads();
  if (tid == 0) {
    for (int c = 0; c < 16; ++c) {
      float d = diag[c * 17 + c];
      for (int m = 0; m < c; ++m) d -= diag[c * 17 + m] * diag[c * 17 + m];
      d = sqrtf(fmaxf(d, 1e-30f));
      diag[c * 17 + c] = d;
      float inv = 1.0f / d;
      for (int r = c + 1; r < 16; ++r) {
        float s = diag[r * 17 + c];
        for (int m = 0; m < c; ++m) s -= diag[r * 17 + m] * diag[c * 17 + m];
        diag[r * 17 + c] = s * inv;
      }
    }
  }
  __syncthreads();
  {
    int r = tid >> 4, c = tid & 15;
    Ab[(size_t)(j0 + r) * DP + (j0 + c)] = (c <= r) ? diag[r * 17 + c] : 0.0f;
  }
}

// ---------------------------------------------------------------------------
// 1b) Cholesky step: panel TRSM  X = A_ib * Ljj^{-T}. One (wave,half) per
//     16-row tile; grid spans all row tiles of the panel.
// ---------------------------------------------------------------------------
__global__ void __launch_bounds__(256)
chol_trsm_kernel(float* __restrict__ A, int jb) {
  const int b = blockIdx.x;
  float* Ab = A + (size_t)b * DP * DP;
  __shared__ float diag[16 * 17];
  const int tid = threadIdx.x;
  const int j0 = jb * 16;
  {
    int r = tid >> 4, c = tid & 15;
    diag[r * 17 + c] = Ab[(size_t)(j0 + r) * DP + (j0 + c)];
  }
  __syncthreads();
  const int lane = tid & 31;
  const int wave = tid >> 5;
  const int hf = lane >> 4;
  const int l16 = lane & 15;
  const int nrb = NB - 1 - jb;
  const int t = blockIdx.y * 16 + wave * 2 + hf;
  if (t < nrb) {
    const int ib = jb + 1 + t;
    const size_t row = (size_t)(ib * 16 + l16);
    float x[16];
#pragma unroll
    for (int c = 0; c < 16; ++c) x[c] = Ab[row * DP + (j0 + c)];
#pragma unroll
    for (int c = 0; c < 16; ++c) {
      float s = x[c];
      for (int m = 0; m < c; ++m) s -= x[m] * diag[c * 17 + m];
      x[c] = s / diag[c * 17 + c];
    }
#pragma unroll
    for (int c = 0; c < 16; ++c) Ab[row * DP + (j0 + c)] = x[c];
  }
}

// ---------------------------------------------------------------------------
// 1c) Cholesky step: trailing SYRK  C_ik -= P_i * P_k^T via WMMA f32 16x16x4.
//     One wave per 16x16 tile; grid spans all (i>=k) tiles -> full-chip width.
// ---------------------------------------------------------------------------
__global__ void __launch_bounds__(256)
chol_syrk_kernel(float* __restrict__ A, int jb) {
  const int b = blockIdx.x;
  float* Ab = A + (size_t)b * DP * DP;
  const int wave = threadIdx.x >> 5;
  const int lane = threadIdx.x & 31;
  const int hf = lane >> 4;
  const int l16 = lane & 15;
  const int nrb = NB - 1 - jb;
  const int ntiles = nrb * (nrb + 1) / 2;
  const int f = blockIdx.y * 8 + wave;
  if (f >= ntiles) return;  // wave-uniform: EXEC stays all-ones for WMMA
  int i = (int)((sqrtf(8.0f * (float)f + 1.0f) - 1.0f) * 0.5f);
  while ((i + 1) * (i + 2) / 2 <= f) ++i;
  while (i * (i + 1) / 2 > f) --i;
  const int kk = f - i * (i + 1) / 2;
  const int ib = jb + 1 + i;
  const int kb = jb + 1 + kk;
  const int j0 = jb * 16;
  const size_t cbase = (size_t)(ib * 16) * DP + (size_t)kb * 16;
  v8f c;
#pragma unroll
  for (int r = 0; r < 8; ++r)
    c[r] = Ab[cbase + (size_t)(r + 8 * hf) * DP + l16];
#pragma unroll
  for (int k0 = 0; k0 < 16; k0 += 4) {
    const int ka = k0 + 2 * hf;
    v2f a, bf;
    a[0] = -Ab[(size_t)(ib * 16 + l16) * DP + j0 + ka];   // A(M=l16, K)
    a[1] = -Ab[(size_t)(ib * 16 + l16) * DP + j0 + ka + 1];
    bf[0] = Ab[(size_t)(kb * 16 + l16) * DP + j0 + ka];   // B(K, N=l16) = P_k^T
    bf[1] = Ab[(size_t)(kb * 16 + l16) * DP + j0 + ka + 1];
    c = wmma_f32_k4(a, bf, c);
  }
#pragma unroll
  for (int r = 0; r < 8; ++r)
    Ab[cbase + (size_t)(r + 8 * hf) * DP + l16] = c[r];
}

// ---------------------------------------------------------------------------
// 2) Forward solve L * Y = I (Y lower-triangular, padded). One wave per
//    16-column block; GEMM accumulation via WMMA; 16x16 diag solve via LDS.
// ---------------------------------------------------------------------------
__global__ void __launch_bounds__(32)
fwd_solve_kernel(const float* __restrict__ A, float* __restrict__ Y) {
  const int b = blockIdx.x / NB;
  const int cb = blockIdx.x % NB;
  const float* Lb = A + (size_t)b * DP * DP;
  float* Yb = Y + (size_t)b * DP * DP;
  __shared__ float tile[16 * 17];
  __shared__ float dt[16 * 17];
  const int lane = threadIdx.x & 31;
  const int hf = lane >> 4;
  const int l16 = lane & 15;
  const int c0 = cb * 16;
  // Zero strictly-upper tiles of this column block.
  for (int idx = lane; idx < cb * 16 * 16; idx += 32) {
    int r = idx >> 4, cc = idx & 15;
    Yb[(size_t)r * DP + c0 + cc] = 0.0f;
  }
  for (int rb = cb; rb < NB; ++rb) {
    v8f acc;
#pragma unroll
    for (int r = 0; r < 8; ++r)
      acc[r] = ((r + 8 * hf) == l16 && rb == cb) ? 1.0f : 0.0f;
    for (int mb = cb; mb < rb; ++mb) {
      // Overlap L2 latency of the next tiles with the dependent WMMA chain.
      __builtin_prefetch(&Lb[(size_t)(rb * 16 + l16) * DP + (mb + 1) * 16], 0, 3);
      __builtin_prefetch(&Yb[(size_t)((mb + 1) * 16 + l16) * DP + c0], 0, 3);
#pragma unroll
      for (int k0 = 0; k0 < 16; k0 += 4) {
        const int ka = k0 + 2 * hf;
        v2f a, bf;
        a[0] = -Lb[(size_t)(rb * 16 + l16) * DP + mb * 16 + ka];
        a[1] = -Lb[(size_t)(rb * 16 + l16) * DP + mb * 16 + ka + 1];
        bf[0] = Yb[(size_t)(mb * 16 + ka) * DP + c0 + l16];
        bf[1] = Yb[(size_t)(mb * 16 + ka + 1) * DP + c0 + l16];
        acc = wmma_f32_k4(a, bf, acc);
      }
    }
#pragma unroll
    for (int r = 0; r < 8; ++r) {
      tile[(r + 8 * hf) * 17 + l16] = acc[r];
      dt[(r + 8 * hf) * 17 + l16] =
          Lb[(size_t)(rb * 16 + r + 8 * hf) * DP + rb * 16 + l16];
    }
    __syncthreads();
    if (lane < 16) {  // forward substitution, one RHS column per lane
      float x[16];
#pragma unroll
      for (int r = 0; r < 16; ++r) {
        float s = tile[r * 17 + lane];
        for (int m = 0; m < r; ++m) s -= dt[r * 17 + m] * x[m];
        x[r] = s / dt[r * 17 + r];
      }
#pragma unroll
      for (int r = 0; r < 16; ++r)
        Yb[(size_t)(rb * 16 + r) * DP + c0 + lane] = x[r];
    }
    __syncthreads();
  }
}

// ---------------------------------------------------------------------------
// 3) Backward solve L^T * P = Y, in place (P = precision, padded).
// ---------------------------------------------------------------------------
__global__ void __launch_bounds__(32)
bwd_solve_kernel(const float* __restrict__ A, float* __restrict__ Y) {
  const int b = blockIdx.x / NB;
  const int cb = blockIdx.x % NB;
  const float* Lb = A + (size_t)b * DP * DP;
  float* Yb = Y + (size_t)b * DP * DP;
  __shared__ float tile[16 * 17];
  __shared__ float dt[16 * 17];
  const int lane = threadIdx.x & 31;
  const int hf = lane >> 4;
  const int l16 = lane & 15;
  const int c0 = cb * 16;
  for (int rb = NB - 1; rb >= 0; --rb) {
    v8f acc;
#pragma unroll
    for (int r = 0; r < 8; ++r)
      acc[r] = Yb[(size_t)(rb * 16 + r + 8 * hf) * DP + c0 + l16];
    for (int mb = rb + 1; mb < NB; ++mb) {
      __builtin_prefetch(&Lb[(size_t)((mb + 1) * 16 + l16) * DP + rb * 16], 0, 3);
      __builtin_prefetch(&Yb[(size_t)((mb + 1) * 16 + l16) * DP + c0], 0, 3);
#pragma unroll
      for (int k0 = 0; k0 < 16; k0 += 4) {
        const int ka = k0 + 2 * hf;
        v2f a, bf;
        // A = L(mb,rb)^T : elem(M,K) = L[mb*16+K][rb*16+M]
        a[0] = -Lb[(size_t)(mb * 16 + ka) * DP + rb * 16 + l16];
        a[1] = -Lb[(size_t)(mb * 16 + ka + 1) * DP + rb * 16 + l16];
        bf[0] = Yb[(size_t)(mb * 16 + ka) * DP + c0 + l16];
        bf[1] = Yb[(size_t)(mb * 16 + ka + 1) * DP + c0 + l16];
        acc = wmma_f32_k4(a, bf, acc);
      }
    }
#pragma unroll
    for (int r = 0; r < 8; ++r) {
      tile[(r + 8 * hf) * 17 + l16] = acc[r];
      dt[(r + 8 * hf) * 17 + l16] =
          Lb[(size_t)(rb * 16 + r + 8 * hf) * DP + rb * 16 + l16];
    }
    __syncthreads();
    if (lane < 16) {  // backward substitution against L_rr^T
      float x[16];
#pragma unroll
      for (int r = 15; r >= 0; --r) {
        float s = tile[r * 17 + lane];
        for (int m = r + 1; m < 16; ++m) s -= dt[m * 17 + r] * x[m];
        x[r] = s / dt[r * 17 + r];
      }
#pragma unroll
      for (int r = 0; r < 16; ++r)
        Yb[(size_t)(rb * 16 + r) * DP + c0 + lane] = x[r];
    }
    __syncthreads();
  }
}

// ---------------------------------------------------------------------------
// 4) maha + half_logdet per batch: forward solve L s = diff, one wave/batch.
// ---------------------------------------------------------------------------
__global__ void __launch_bounds__(32)
maha_kernel(const float* __restrict__ A, const float* __restrict__ diff,
            float* __restrict__ maha, float* __restrict__ hld) {
  const int b = blockIdx.x;
  const float* Lb = A + (size_t)b * DP * DP;
  const float* db = diff + (size_t)b * DD;
  __shared__ float s[DD];
  const int lane = threadIdx.x & 31;
  float am = 0.0f, al = 0.0f;
  for (int r = 0; r < DD; ++r) {
    const float* Lr = Lb + (size_t)r * DP;
    float part = 0.0f;
    for (int m = lane; m < r; m += 32) part += Lr[m] * s[m];
    part = wave_reduce_sum(part);
    const float lrr = Lr[r];
    const float sr = (db[r] - part) / lrr;
    if (lane == 0) {
      s[r] = sr;
      am += sr * sr;
      al += logf(lrr);
    }
    __syncthreads();
  }
  if (lane == 0) {
    maha[b] = am;
    hld[b] = al;
  }
}

// ---------------------------------------------------------------------------
// 5) slogdet of the K spatial (207x207) covariances: lane-parallel Cholesky.
// ---------------------------------------------------------------------------
__global__ void __launch_bounds__(32)
covs_logdet_kernel(const float* __restrict__ covs, float* __restrict__ work,
                   float* __restrict__ ld_s) {
  const int k = blockIdx.x;
  const float* Sin = covs + (size_t)k * NN * NN;
  float* S = work + (size_t)k * NN * NN;
  const int lane = threadIdx.x & 31;
  for (int idx = lane; idx < NN * NN; idx += 32) S[idx] = Sin[idx];
  __syncthreads();
  float al = 0.0f;
  for (int j = 0; j < NN; ++j) {
    float part = 0.0f;
    for (int m = lane; m < j; m += 32) {
      float v = S[j * NN + m];
      part += v * v;
    }
    part = wave_reduce_sum(part);
    const float djj = sqrtf(fmaxf(S[j * NN + j] - part, 1e-30f));
    if (lane == 0) {
      S[j * NN + j] = djj;
      al += logf(djj);
    }
    for (int i = j + 1 + lane; i < NN; i += 32) {
      float sacc = S[i * NN + j];
      for (int m = 0; m < j; ++m) sacc -= S[i * NN + m] * S[j * NN + m];
      S[i * NN + j] = sacc / djj;
    }
    __syncthreads();
  }
  if (lane == 0) ld_s[k] = 2.0f * al;
}

// ---------------------------------------------------------------------------
// 6) slogdet of the K temporal (12x12) covariances (tiny, sequential).
// ---------------------------------------------------------------------------
__global__ void __launch_bounds__(32)
covt_logdet_kernel(const float* __restrict__ covt, float* __restrict__ ld_t) {
  const int k = blockIdx.x;
  if (threadIdx.x != 0) return;
  float Lm[TT][TT];
  for (int i = 0; i < TT; ++i)
    for (int j = 0; j < TT; ++j)
      Lm[i][j] = covt[(size_t)k * TT * TT + i * TT + j];
  float al = 0.0f;
  for (int c = 0; c < TT; ++c) {
    float d = Lm[c][c];
    for (int m = 0; m < c; ++m) d -= Lm[c][m] * Lm[c][m];
    d = sqrtf(fmaxf(d, 1e-30f));
    Lm[c][c] = d;
    al += logf(d);
    for (int r = c + 1; r < TT; ++r) {
      float sv = Lm[r][c];
      for (int m = 0; m < c; ++m) sv -= Lm[r][m] * Lm[c][m];
      Lm[r][c] = sv / d;
    }
  }
  ld_t[k] = 2.0f * al;
}

// ---------------------------------------------------------------------------
// 7) Kronecker quadratic form: quad[b][k] = sum_ij covs[k][i][j] z_i Ct_k z_j^T
// ---------------------------------------------------------------------------
__global__ void __launch_bounds__(256)
kron_kernel(const float* __restrict__ diff, const float* __restrict__ covs,
            const float* __restrict__ covt, float* __restrict__ quad) {
  const int b = blockIdx.x / KK;
  const int k = blockIdx.x % KK;
  __shared__ float z[DD];
  __shared__ float W[DD];
  __shared__ float ct[TT * TT];
  __shared__ float red[32];
  const int tid = threadIdx.x;
  for (int idx = tid; idx < DD; idx += 256) z[idx] = diff[(size_t)b * DD + idx];
  if (tid < TT * TT) ct[tid] = covt[(size_t)k * TT * TT + tid];
  __syncthreads();
  for (int idx = tid; idx < DD; idx += 256) {
    const int j = idx / TT, t = idx % TT;
    float sacc = 0.0f;
#pragma unroll
    for (int u = 0; u < TT; ++u) sacc += ct[t * TT + u] * z[j * TT + u];
    W[idx] = sacc;
  }
  __syncthreads();
  float part = 0.0f;
  for (int i = tid; i < NN; i += 256) {
    float g[TT];
#pragma unroll
    for (int t = 0; t < TT; ++t) g[t] = 0.0f;
    const float* cr = covs + ((size_t)k * NN + i) * NN;
    for (int j = 0; j < NN; ++j) {
      const float cv = cr[j];
#pragma unroll
      for (int t = 0; t < TT; ++t) g[t] += cv * W[j * TT + t];
    }
#pragma unroll
    for (int t = 0; t < TT; ++t) part += z[i * TT + t] * g[t];
  }
  part = block_reduce_sum(part, red);
  if (tid == 0) quad[b * KK + k] = part;
}

// ---------------------------------------------------------------------------
// 8) reg: sum |offdiag(precision)| over the unpadded D x D region.
// ---------------------------------------------------------------------------
__global__ void __launch_bounds__(256)
reg_reduce_kernel(const float* __restrict__ P, float* __restrict__ regsum) {
  __shared__ float red[32];
  const size_t total = (size_t)BB * DD * DD;
  float part = 0.0f;
  for (size_t idx = (size_t)blockIdx.x * 256u + threadIdx.x; idx < total;
       idx += (size_t)gridDim.x * 256u) {
    const size_t b = idx / ((size_t)DD * DD);
    const size_t rem = idx % ((size_t)DD * DD);
    const size_t r = rem / DD, c = rem % DD;
    if (r != c) part += fabsf(P[b * (size_t)DP * DP + r * DP + c]);
  }
  part = block_reduce_sum(part, red);
  if (threadIdx.x == 0) atomicAdd(regsum, part);
}

// ---------------------------------------------------------------------------
// 9) Combine: mse + nll + reg + per-sample Kronecker nll -> out[8].
// ---------------------------------------------------------------------------
__global__ void __launch_bounds__(256)
combine_kernel(const float* __restrict__ y, const float* __restrict__ w,
               const float* __restrict__ mu, const float* __restrict__ maha,
               const float* __restrict__ hld, const float* __restrict__ quad,
               const float* __restrict__ ld_s, const float* __restrict__ ld_t,
               const float* __restrict__ regsum, float* __restrict__ out) {
  __shared__ float red[32];
  float part = 0.0f;
  for (int idx = threadIdx.x; idx < BB * DD; idx += 256) {
    const float d = mu[idx] - y[idx];
    part += d * d;
  }
  part = block_reduce_sum(part, red);
  if (threadIdx.x == 0) {
    const float LOG2PI = 1.8378770664093453f;
    const float mse = part / (float)(BB * DD);
    float nll = 0.0f;
    for (int b = 0; b < BB; ++b) {
      const float lp = -0.5f * ((float)DD * LOG2PI + maha[b]) - hld[b];
      nll -= lp;
    }
    nll /= (float)BB;
    const float reg = regsum[0] / ((float)BB * (float)DD * (float)(DD - 1));
    float sum_ld = 0.0f;
    for (int k = 0; k < KK; ++k)
      sum_ld += (float)NN * ld_t[k] + (float)TT * ld_s[k];
    out[0] = 0.1f * nll + 0.1f * reg + 0.1f * mse;
    out[1] = nll;
    out[2] = reg;
    out[3] = mse;
    for (int b = 0; b < BB; ++b) {
      float s1 = 0.0f, s2 = 0.0f;
      for (int k = 0; k < KK; ++k) {
        s1 += w[b * KK + k] * quad[b * KK + k];
        s2 += logf(w[b * KK + k]);
      }
      out[4 + b] = -0.5f * s1 + s2 + sum_ld;
    }
  }
}

extern "C" void kernel_launch(void* const* d_in, const int* in_sizes, int n_in,
                              void* d_out, int out_size, void* d_ws,
                              size_t ws_size, hipStream_t stream) {
  (void)in_sizes; (void)n_in; (void)out_size; (void)ws_size;
  const float* y = (const float*)d_in[0];
  const float* w = (const float*)d_in[1];
  const float* mu = (const float*)d_in[2];
  const float* covs = (const float*)d_in[3];
  const float* covt = (const float*)d_in[4];
  const float* cov = (const float*)d_in[5];
  float* out = (float*)d_out;

  float* ws = (float*)d_ws;
  const size_t matsz = (size_t)BB * DP * DP;
  float* Apad = ws;                 // B*DP*DP  : padded cov -> L
  float* Yw = Apad + matsz;         // B*DP*DP  : L^{-1} then precision
  float* diff = Yw + matsz;         // B*D
  float* mahav = diff + (size_t)BB * DD;  // B
  float* hldv = mahav + BB;         // B
  float* quadv = hldv + BB;         // B*K
  float* ldsv = quadv + BB * KK;    // K
  float* ldtv = ldsv + KK;          // K
  float* regsum = ldtv + KK;        // 1
  float* swork = regsum + 1;        // K*NN*NN

  prep_kernel<<<4096, 256, 0, stream>>>(cov, y, mu, Apad, diff, regsum);

  // Blocked Cholesky, one panel step per launch trio; SYRK spans the chip.
  for (int jb = 0; jb < NB; ++jb) {
    chol_diag_kernel<<<BB, 256, 0, stream>>>(Apad, jb);
    const int nrb = NB - 1 - jb;
    if (nrb > 0) {
      dim3 gt(BB, (unsigned)((nrb + 15) / 16));
      chol_trsm_kernel<<<gt, 256, 0, stream>>>(Apad, jb);
      const int ntiles = nrb * (nrb + 1) / 2;
      dim3 gs(BB, (unsigned)((ntiles + 7) / 8));
      chol_syrk_kernel<<<gs, 256, 0, stream>>>(Apad, jb);
    }
  }

  fwd_solve_kernel<<<BB * NB, 32, 0, stream>>>(Apad, Yw);
  bwd_solve_kernel<<<BB * NB, 32, 0, stream>>>(Apad, Yw);
  maha_kernel<<<BB, 32, 0, stream>>>(Apad, diff, mahav, hldv);
  covs_logdet_kernel<<<KK, 32, 0, stream>>>(covs, swork, ldsv);
  covt_logdet_kernel<<<KK, 32, 0, stream>>>(covt, ldtv);
  kron_kernel<<<BB * KK, 256, 0, stream>>>(diff, covs, covt, quadv);
  reg_reduce_kernel<<<2048, 256, 0, stream>>>(Yw, regsum);
  combine_kernel<<<1, 256, 0, stream>>>(y, w, mu, mahav, hldv, quadv, ldsv,
                                        ldtv, regsum, out);
}